// MaxUnpooling2D_64415919505944
// MI455X (gfx1250) — compile-verified
//
#include <hip/hip_runtime.h>

// Max-unpool scatter for MI455X (gfx1250), compile-only tuned.
//
// Pure streaming problem: 32 MB x + 32 MB idx in, 128 MB out => ~192 MB of
// traffic, ~8.2 us at 23.3 TB/s. No matrix math exists in this op, so the
// optimization targets are data movement only:
//   * fused zero-fill + scatter (each pooled element owns its 2x2 output
//     block cells -> write value to the matching cell, 0 to the other 3;
//     no atomics, no memset pass, full output coverage, deterministic)
//   * B128 loads/stores, contiguous per wave32
//   * non-temporal cache policy (working set == 192 MB L2, zero reuse)
//   * global_prefetch_b8 of the next grid-stride tile

typedef float v4f __attribute__((ext_vector_type(4)));
typedef int   v4i __attribute__((ext_vector_type(4)));

namespace {
constexpr int kB  = 16, kH = 64, kW = 64, kC = 128;
constexpr int kH2 = 2 * kH, kW2 = 2 * kW;
constexpr int kGroups = kB * kH * kW * (kC / 4);   // 2,097,152 float4 groups
constexpr int kBlock  = 256;                       // 8 waves (wave32) per WG
constexpr int kGrid   = 2048;                      // -> 4 groups per thread
}

__global__ __launch_bounds__(kBlock) void unpool_scatter_kernel(
    const float* __restrict__ x,
    const int*   __restrict__ index,
    float*       __restrict__ out)
{
  const int stride = kGrid * kBlock;
  for (int i = (int)blockIdx.x * kBlock + (int)threadIdx.x; i < kGroups;
       i += stride) {
    // Prefetch next grid-stride iteration (emits global_prefetch_b8).
    const int nxt = i + stride;
    if (nxt < kGroups) {
      __builtin_prefetch((const v4f*)x + nxt, 0, 1);
      __builtin_prefetch((const v4i*)index + nxt, 0, 1);
    }

    // Streaming non-temporal B128 loads: 4 channels per thread.
    const v4f xv = __builtin_nontemporal_load((const v4f*)x + i);
    const v4i iv = __builtin_nontemporal_load((const v4i*)index + i);

    // Decompose linear group id -> (b, h, w, channel-group).
    const int cg = i & (kC / 4 - 1);   // 32 channel groups of 4
    const int r  = i >> 5;
    const int w  = r & (kW - 1);
    const int r2 = r >> 6;
    const int h  = r2 & (kH - 1);
    const int b  = r2 >> 6;
    const int c  = cg << 2;

    // Flat output indices of the 4 cells of this element's 2x2 block
    // (at base channel c). Max index 33,554,431 fits in int32.
    const int a0 = ((b * kH2 + 2 * h) * kW2 + 2 * w) * kC + c;  // (2h  ,2w  )
    const int a1 = a0 + kC;                                     // (2h  ,2w+1)
    const int a2 = a0 + kW2 * kC;                               // (2h+1,2w  )
    const int a3 = a2 + kC;                                     // (2h+1,2w+1)

    const int cells[4] = {a0, a1, a2, a3};
#pragma unroll
    for (int k = 0; k < 4; ++k) {
      const int a = cells[k];
      // Branch-free select: the index generator guarantees each idx lands in
      // one of these 4 cells, so value goes to the match, zero elsewhere.
      v4f o;
      o.x = (iv.x == a + 0) ? xv.x : 0.0f;
      o.y = (iv.y == a + 1) ? xv.y : 0.0f;
      o.z = (iv.z == a + 2) ? xv.z : 0.0f;
      o.w = (iv.w == a + 3) ? xv.w : 0.0f;
      // Per wave32, each cell store is a contiguous 512 B NT B128 burst.
      __builtin_nontemporal_store(o, (v4f*)(out + a));
    }
  }
}

extern "C" void kernel_launch(void* const* d_in, const int* in_sizes, int n_in,
                              void* d_out, int out_size, void* d_ws,
                              size_t ws_size, hipStream_t stream) {
  const float* x     = (const float*)d_in[0];
  const int*   index = (const int*)d_in[1];
  float*       out   = (float*)d_out;
  (void)in_sizes; (void)n_in; (void)out_size; (void)d_ws; (void)ws_size;
  unpool_scatter_kernel<<<kGrid, kBlock, 0, stream>>>(x, index, out);
}